// CrossWindowAttention_712964571284
// MI455X (gfx1250) — compile-verified
//
#include <hip/hip_runtime.h>
#include <math.h>

typedef float v2f __attribute__((ext_vector_type(2)));
typedef float v8f __attribute__((ext_vector_type(8)));

#define NTOK 64
#define DIM 128
#define HEADS 4
#define HD 32
#define RATIO 4

// LDS pitches (floats), chosen for bank-conflict-free WMMA fragment access
#define XP 132   // x tile / xb pitch
#define QP 36    // q (64 x 32)
#define KP 72    // k^T (32 x 64)
#define VP 40    // v (64 x 32)
#define SP 68    // S / P (64 x 64), aliased onto x tile

__device__ __forceinline__ v8f wmma4(v2f a, v2f b, v8f c) {
  // D(16x16) = A(16x4) * B(4x16) + C, fp32
  return __builtin_amdgcn_wmma_f32_16x16x4_f32(false, a, false, b, (short)0, c,
                                               false, false);
}

__global__ __launch_bounds__(256) void attn_kernel(
    const float* __restrict__ x1, const float* __restrict__ x2,
    const float* __restrict__ w1, const float* __restrict__ b1v,
    const float* __restrict__ w2, const float* __restrict__ b2v,
    const float* __restrict__ rpb, const int* __restrict__ ridx,
    float* __restrict__ attn_out, float* __restrict__ xacc, int B1) {
  __shared__ __align__(16) float x_tile[NTOK * XP];  // aliases s_buf (64*SP <= 64*XP)
  __shared__ __align__(16) float q_lds[NTOK * QP];
  __shared__ __align__(16) float kT_lds[HD * KP];
  __shared__ __align__(16) float v_lds[NTOK * VP];
  float* s_buf = x_tile;

  const int tid = threadIdx.x;
  const int lane = tid & 31;
  const int w = tid >> 5;        // wave id 0..7
  const int lmod = lane & 15;    // 16-dim index within fragment
  const int lhi = lane >> 4;     // half-wave: K-split / M-split selector
  const int b = blockIdx.x / HEADS;
  const int h = blockIdx.x % HEADS;

  const int mt = w >> 1;  // 0..3 : M tile for q/k/v/O
  const int nt = w & 1;   // 0..1 : N tile for q/k/v/O
  const float scale = 0.17677669529663687f;  // 32^-0.5

  v8f accO;
  for (int i = 0; i < 8; ++i) accO[i] = 0.f;

  for (int r = 0; r < RATIO; ++r) {
    const int b2i = RATIO * b + r;
    const int b1i = b2i & (B1 - 1);  // q.repeat: q index = b2 mod B1 (B1 = 2048)
    __syncthreads();
    // ---- load x1 window (64x128) ----
    {
      const float* src = x1 + (size_t)b1i * (NTOK * DIM);
      for (int i = 0; i < 8; ++i) {
        int j = tid + 256 * i;  // float4 units
        int row = j >> 5, c4 = (j & 31) << 2;
        const float4 d = *(const float4*)(src + row * DIM + c4);
        float* dst = &x_tile[row * XP + c4];
        dst[0] = d.x; dst[1] = d.y; dst[2] = d.z; dst[3] = d.w;
      }
    }
    __syncthreads();
    // ---- q = (x1 @ Wq^T + b) * scale, head slice ----
    {
      v8f c;
      for (int i = 0; i < 8; ++i) c[i] = 0.f;
      const int wrow = h * HD + nt * 16 + lmod;
      const float* wp = w1 + wrow * DIM;
      #pragma unroll 8
      for (int kk = 0; kk < 32; ++kk) {
        const int k0 = kk * 4 + 2 * lhi;
        v2f a = *(const v2f*)&x_tile[(mt * 16 + lmod) * XP + k0];
        v2f bb = *(const v2f*)(wp + k0);  // B[k][n] = Wq[n][k], contiguous pair
        c = wmma4(a, bb, c);
      }
      const float bias = b1v[wrow];
      #pragma unroll
      for (int i = 0; i < 8; ++i) {
        int m = mt * 16 + i + 8 * lhi;
        q_lds[m * QP + nt * 16 + lmod] = (c[i] + bias) * scale;
      }
    }
    __syncthreads();
    // ---- load x2 window (overwrites x_tile) ----
    {
      const float* src = x2 + (size_t)b2i * (NTOK * DIM);
      for (int i = 0; i < 8; ++i) {
        int j = tid + 256 * i;
        int row = j >> 5, c4 = (j & 31) << 2;
        const float4 d = *(const float4*)(src + row * DIM + c4);
        float* dst = &x_tile[row * XP + c4];
        dst[0] = d.x; dst[1] = d.y; dst[2] = d.z; dst[3] = d.w;
      }
    }
    __syncthreads();
    // ---- k, v projections (head slice); k stored transposed ----
    {
      v8f ck, cv;
      for (int i = 0; i < 8; ++i) { ck[i] = 0.f; cv[i] = 0.f; }
      const int dd = nt * 16 + lmod;
      const float* wkp = w2 + (size_t)(h * HD + dd) * DIM;
      const float* wvp = w2 + (size_t)(DIM + h * HD + dd) * DIM;
      #pragma unroll 4
      for (int kk = 0; kk < 32; ++kk) {
        const int k0 = kk * 4 + 2 * lhi;
        v2f a = *(const v2f*)&x_tile[(mt * 16 + lmod) * XP + k0];
        v2f bk = *(const v2f*)(wkp + k0);
        v2f bv = *(const v2f*)(wvp + k0);
        ck = wmma4(a, bk, ck);
        cv = wmma4(a, bv, cv);
      }
      const float bk0 = b2v[h * HD + dd];
      const float bv0 = b2v[DIM + h * HD + dd];
      #pragma unroll
      for (int i = 0; i < 8; ++i) {
        int tok = mt * 16 + i + 8 * lhi;
        kT_lds[dd * KP + tok] = ck[i] + bk0;
        v_lds[tok * VP + dd] = cv[i] + bv0;
      }
    }
    __syncthreads();
    // ---- S = q @ k^T + rel-pos bias -> s_buf (aliases x_tile, now dead) ----
    for (int t = 2 * w; t < 2 * w + 2; ++t) {
      const int mts = t >> 2, nts = t & 3;
      v8f c;
      for (int i = 0; i < 8; ++i) c[i] = 0.f;
      #pragma unroll
      for (int kk = 0; kk < 8; ++kk) {
        const int k0 = kk * 4 + 2 * lhi;
        v2f a = *(const v2f*)&q_lds[(mts * 16 + lmod) * QP + k0];
        v2f bb;
        bb.x = kT_lds[k0 * KP + nts * 16 + lmod];
        bb.y = kT_lds[(k0 + 1) * KP + nts * 16 + lmod];
        c = wmma4(a, bb, c);
      }
      #pragma unroll
      for (int i = 0; i < 8; ++i) {
        int m = mts * 16 + i + 8 * lhi;
        int n = nts * 16 + lmod;
        int bi = ridx[m * NTOK + n];
        s_buf[m * SP + n] = c[i] + rpb[bi * HEADS + h];
      }
    }
    __syncthreads();
    // ---- softmax rows (4 lanes per row) + write attn ----
    {
      const int row = tid >> 2;
      const int part = tid & 3;
      float vals[16];
      const float* sp = &s_buf[row * SP + part * 16];
      float mx = -INFINITY;
      #pragma unroll
      for (int i = 0; i < 16; ++i) { vals[i] = sp[i]; mx = fmaxf(mx, vals[i]); }
      mx = fmaxf(mx, __shfl_xor(mx, 1, 4));
      mx = fmaxf(mx, __shfl_xor(mx, 2, 4));
      float sum = 0.f;
      #pragma unroll
      for (int i = 0; i < 16; ++i) {
        float e = __expf(vals[i] - mx);
        vals[i] = e;
        sum += e;
      }
      sum += __shfl_xor(sum, 1, 4);
      sum += __shfl_xor(sum, 2, 4);
      const float inv = 1.0f / sum;
      float* sw = &s_buf[row * SP + part * 16];
      float4* gout = (float4*)(attn_out +
                               ((size_t)(b2i * HEADS + h) * NTOK + row) * NTOK +
                               part * 16);
      #pragma unroll
      for (int i4 = 0; i4 < 4; ++i4) {
        float4 pv;
        pv.x = vals[4 * i4 + 0] * inv;
        pv.y = vals[4 * i4 + 1] * inv;
        pv.z = vals[4 * i4 + 2] * inv;
        pv.w = vals[4 * i4 + 3] * inv;
        sw[4 * i4 + 0] = pv.x; sw[4 * i4 + 1] = pv.y;
        sw[4 * i4 + 2] = pv.z; sw[4 * i4 + 3] = pv.w;
        gout[i4] = pv;
      }
    }
    __syncthreads();
    // ---- O += P @ v (accumulated across r in registers) ----
    {
      #pragma unroll 4
      for (int kk = 0; kk < 16; ++kk) {
        const int k0 = kk * 4 + 2 * lhi;
        v2f a = *(const v2f*)&s_buf[(mt * 16 + lmod) * SP + k0];
        v2f bb;
        bb.x = v_lds[k0 * VP + nt * 16 + lmod];
        bb.y = v_lds[(k0 + 1) * VP + nt * 16 + lmod];
        accO = wmma4(a, bb, accO);
      }
    }
  }
  // ---- write pre-projection accumulator: xacc[b][tok][h*32+dd] ----
  {
    const int dd = nt * 16 + lmod;
    #pragma unroll
    for (int i = 0; i < 8; ++i) {
      int m = mt * 16 + i + 8 * lhi;
      xacc[((size_t)b * NTOK + m) * DIM + h * HD + dd] = accO[i];
    }
  }
}

__global__ __launch_bounds__(256) void proj_kernel(
    const float* __restrict__ xacc, const float* __restrict__ pw,
    const float* __restrict__ pb, float* __restrict__ out) {
  __shared__ __align__(16) float xb[NTOK * XP];
  const int tid = threadIdx.x, lane = tid & 31, w = tid >> 5;
  const int lmod = lane & 15, lhi = lane >> 4;
  const int b = blockIdx.x;
  const float* src = xacc + (size_t)b * (NTOK * DIM);
  for (int i = 0; i < 8; ++i) {
    int j = tid + 256 * i;
    int row = j >> 5, c4 = (j & 31) << 2;
    const float4 d = *(const float4*)(src + row * DIM + c4);
    float* dst = &xb[row * XP + c4];
    dst[0] = d.x; dst[1] = d.y; dst[2] = d.z; dst[3] = d.w;
  }
  __syncthreads();
  const int n = w * 16 + lmod;  // wave = N tile (8 tiles of 16)
  const float* wp = pw + (size_t)n * DIM;
  const float bias = pb[n];
  for (int mt2 = 0; mt2 < 4; ++mt2) {
    v8f c;
    for (int i = 0; i < 8; ++i) c[i] = 0.f;
    #pragma unroll 8
    for (int kk = 0; kk < 32; ++kk) {
      const int k0 = kk * 4 + 2 * lhi;
      v2f a = *(const v2f*)&xb[(mt2 * 16 + lmod) * XP + k0];
      v2f bb = *(const v2f*)(wp + k0);
      c = wmma4(a, bb, c);
    }
    #pragma unroll
    for (int i = 0; i < 8; ++i) {
      int m = mt2 * 16 + i + 8 * lhi;
      out[((size_t)b * NTOK + m) * DIM + n] = c[i] + bias;
    }
  }
}

extern "C" void kernel_launch(void* const* d_in, const int* in_sizes, int n_in,
                              void* d_out, int out_size, void* d_ws,
                              size_t ws_size, hipStream_t stream) {
  const float* x1 = (const float*)d_in[0];
  const float* x2 = (const float*)d_in[1];
  const float* w1 = (const float*)d_in[2];
  const float* b1 = (const float*)d_in[3];
  const float* w2 = (const float*)d_in[4];
  const float* b2 = (const float*)d_in[5];
  const float* pw = (const float*)d_in[6];
  const float* pb = (const float*)d_in[7];
  const float* rpb = (const float*)d_in[8];
  const int* ridx = (const int*)d_in[9];
  const int B1 = in_sizes[0] / (NTOK * DIM);  // 2048

  float* xout = (float*)d_out;
  float* attn = xout + (size_t)B1 * NTOK * DIM;  // tuple order: (x, attn)
  float* xacc = (float*)d_ws;                    // (B1, 64, 128) fp32 scratch

  attn_kernel<<<dim3(B1 * HEADS), dim3(256), 0, stream>>>(
      x1, x2, w1, b1, w2, b2, rpb, ridx, attn, xacc, B1);
  proj_kernel<<<dim3(B1), dim3(256), 0, stream>>>(xacc, pw, pb, xout);
}